// ModulationIntegrator_32169305047592
// MI455X (gfx1250) — compile-verified
//
#include <hip/hip_runtime.h>

typedef __attribute__((ext_vector_type(2))) float v2f;
typedef __attribute__((ext_vector_type(8))) float v8f;

#define BB      64
#define RR      16
#define NBLK    19
#define PP      12
#define KK      4096
#define NTOK    16384
#define NTXT    512
#define SEG     2048              // tokens per segment (LDS counters = 16*2048*4 = 128 KB)
#define NSEG    (NTOK / SEG)      // 8 segments per batch
#define THREADS 256

// One workgroup = (batch b, token segment seg).
// Phase 1: LDS histogram cnt[r][t_local] over this batch's 65536 indices.
// Phase 2: per 16-token tile, WMMA f32 16x16x4 chain (K=16 r-slots) against the
//          16x16 params matrix (cols >= 6 are zero), store 6 valid columns.
__global__ __launch_bounds__(THREADS)
void modint_hist_wmma(const float* __restrict__ adaln,
                      const int*   __restrict__ img_idxs,
                      const int*   __restrict__ block_idx_p,
                      float*       __restrict__ out)
{
    extern __shared__ unsigned cnt[];   // [RR][SEG] u32, layout r*SEG + t (bank-friendly)

    const int tid = threadIdx.x;
    const int b   = blockIdx.x / NSEG;
    const int seg = blockIdx.x % NSEG;
    const int bi  = block_idx_p[0];     // == 7 in this instance

    // ---- zero counters (128 KB, vectorized) ----
    {
        uint4* c4 = (uint4*)cnt;
        const int n4 = (RR * SEG) / 4;
        uint4 z; z.x = z.y = z.z = z.w = 0u;
        for (int i = tid; i < n4; i += THREADS) c4[i] = z;
    }

    // ---- per-lane B fragments: params matrix P[16 r][16 p], p>=6 -> 0 ----
    // B 4x16 layout: VGPR0 = rows {4c+0 | 4c+2} across lane halves, VGPR1 = {4c+1 | 4c+3}.
    const int lane = tid & 31;
    const int hi   = lane >> 4;         // lane half
    const int p    = lane & 15;         // N coordinate
    float bx[4], by[4];
    #pragma unroll
    for (int c = 0; c < 4; ++c) {
        const int r0 = 4 * c + 2 * hi;
        float vx = 0.0f, vy = 0.0f;
        if (p < 6) {
            vx = adaln[((size_t)(b * RR + r0)     * NBLK + bi) * PP + p];
            vy = adaln[((size_t)(b * RR + r0 + 1) * NBLK + bi) * PP + p];
        }
        bx[c] = vx;
        by[c] = vy;
    }

    __syncthreads();

    // ---- histogram ----
    {
        const int4* idx4 = (const int4*)(img_idxs + (size_t)b * (RR * KK));
        const int n = (RR * KK) / 4;             // 16384 int4 loads across the block
        for (int i = tid; i < n; i += THREADS) {
            int4 v = idx4[i];
            const int r = (i * 4) >> 12;         // element_index / 4096 (same r for all 4)
            const unsigned rbase = (unsigned)r * SEG;
            if ((v.x >> 11) == seg) atomicAdd(&cnt[rbase + (v.x & (SEG - 1))], 1u);
            if ((v.y >> 11) == seg) atomicAdd(&cnt[rbase + (v.y & (SEG - 1))], 1u);
            if ((v.z >> 11) == seg) atomicAdd(&cnt[rbase + (v.z & (SEG - 1))], 1u);
            if ((v.w >> 11) == seg) atomicAdd(&cnt[rbase + (v.w & (SEG - 1))], 1u);
        }
    }

    __syncthreads();

    // ---- GEMM: out_tile[16 tok x 16 p] = counts[16 tok x 16 r] @ P[16 r x 16 p] ----
    const int wave     = tid >> 5;
    const int lo       = lane & 15;              // M coordinate within tile (for A)
    const int seg_base = seg * SEG;
    for (int tile = wave; tile < SEG / 16; tile += THREADS / 32) {
        const int t0 = tile * 16;
        v8f acc = {};
        #pragma unroll
        for (int c = 0; c < 4; ++c) {
            // A 16x4 layout: VGPR0 = K={4c+0 | 4c+2}, VGPR1 = K={4c+1 | 4c+3}
            const int k0 = 4 * c + 2 * hi;
            v2f a;
            a.x = (float)cnt[(unsigned)k0 * SEG + t0 + lo];
            a.y = (float)cnt[(unsigned)(k0 + 1) * SEG + t0 + lo];
            v2f bf; bf.x = bx[c]; bf.y = by[c];
            acc = __builtin_amdgcn_wmma_f32_16x16x4_f32(
                false, a, false, bf, (short)0, acc, false, false);
        }
        // D layout: VGPR j -> (M = j + 8*hi, N = lo); only N < 6 is a real output
        if (p < 6) {
            #pragma unroll
            for (int j = 0; j < 8; ++j) {
                const int t = seg_base + t0 + j + 8 * hi;
                out[((size_t)b * NTOK + t) * 6 + p] = acc[j];
            }
        }
    }
}

// txt_offsets = zeros(B, 512, 6); d_out is poisoned, so write it every call.
__global__ __launch_bounds__(THREADS)
void zero_txt(float* __restrict__ out)
{
    const size_t base = (size_t)BB * NTOK * 6;        // start of txt block (float index)
    const size_t n4   = ((size_t)BB * NTXT * 6) / 4;  // 49152 float4
    const size_t i    = (size_t)blockIdx.x * blockDim.x + threadIdx.x;
    if (i < n4) {
        float4* dst = (float4*)(out + base);
        float4 z; z.x = z.y = z.z = z.w = 0.0f;
        dst[i] = z;
    }
}

extern "C" void kernel_launch(void* const* d_in, const int* in_sizes, int n_in,
                              void* d_out, int out_size, void* d_ws, size_t ws_size,
                              hipStream_t stream)
{
    const float* adaln     = (const float*)d_in[0];
    // d_in[1] = alpha : unused by the reference computation
    const int*   img_idxs  = (const int*)d_in[2];
    const int*   block_idx = (const int*)d_in[5];
    float*       out       = (float*)d_out;

    const size_t lds_bytes = (size_t)RR * SEG * sizeof(unsigned);   // 128 KB
    modint_hist_wmma<<<dim3(BB * NSEG), dim3(THREADS), lds_bytes, stream>>>(
        adaln, img_idxs, block_idx, out);

    const int n4 = (BB * NTXT * 6) / 4;                             // 49152
    zero_txt<<<dim3((n4 + THREADS - 1) / THREADS), dim3(THREADS), 0, stream>>>(out);
}